// TauLoss_29918742184749
// MI455X (gfx1250) — compile-verified
//
#include <hip/hip_runtime.h>
#include <hip/hip_bf16.h>

// Soft Kendall-tau loss:
//   out = 1 - (1/(C*n*(n-1))) * sum_c sum_{i,j} tanh((p[c,i]-p[c,j]) * sign(l[c,i]-l[c,j]))
// with C = 128 rows, n = 1024 columns (reference transposes, which maps to
// per-row pairwise sums over the original [128,1024] layout).
//
// Strategy (MI455X / gfx1250, wave32):
//  * Inputs are 1 MB total -> L2-resident; kernel is TRANS(v_tanh_f32)/VALU bound.
//  * tanh(d*sgn) == sgn*tanh(d) for sgn in {-1,0,1}: one tanh + 2 cndmask per pair.
//  * Block = (row c, strip of 256 j's). 256 threads each hold 4 (p_i,l_i) in
//    registers (whole row register-resident per block); stream j-strip from LDS
//    with broadcast reads. 512 blocks x 8 wave32 = 4096 waves.
//  * Final 512-element reduction uses V_WMMA_F32_16X16X4_F32 with an all-ones
//    B matrix (exact f32 math) to collapse 32 lane sums, then lane shuffles.

#ifndef TAU_N
#define TAU_N 1024
#define TAU_C 128
#define TAU_JSTRIPS 4          // 1024/256 j's per block
#define TAU_TPB 256
#define TAU_IPT 4              // i's per thread: 1024 / 256
#endif

typedef __attribute__((ext_vector_type(2))) float v2f;
typedef __attribute__((ext_vector_type(8))) float v8f;

__device__ __forceinline__ float fast_tanhf(float x) {
#if __has_builtin(__builtin_amdgcn_tanhf)
    return __builtin_amdgcn_tanhf(x);     // v_tanh_f32 (gfx1250 trans op)
#else
    return tanhf(x);
#endif
}

// ---------------------------------------------------------------------------
// Kernel 1: per-(row, j-strip) partial pairwise sum.
// grid = (C, JSTRIPS), block = 256 threads.
// ---------------------------------------------------------------------------
__global__ __launch_bounds__(TAU_TPB)
void tau_partial_kernel(const float* __restrict__ pred,
                        const float* __restrict__ y,
                        float* __restrict__ partial) {
    constexpr int N = TAU_N;
    const int c   = blockIdx.x;          // row 0..127
    const int js  = blockIdx.y;          // j-strip 0..3
    const int tid = threadIdx.x;         // 0..255

    const float* __restrict__ prow = pred + (size_t)c * N;
    const float* __restrict__ lrow = y    + (size_t)c * N;

    // Stage this block's j-strip (256 elements of p and l) in LDS.
    __shared__ float sp[TAU_TPB];
    __shared__ float sl[TAU_TPB];
    const int j0 = js * TAU_TPB;
    sp[tid] = prow[j0 + tid];
    sl[tid] = lrow[j0 + tid];

    // Register-cache this thread's TAU_IPT i-values (coalesced global loads).
    float pi[TAU_IPT], li[TAU_IPT], acc[TAU_IPT];
#pragma unroll
    for (int r = 0; r < TAU_IPT; ++r) {
        const int i = tid + r * TAU_TPB;
        pi[r]  = prow[i];
        li[r]  = lrow[i];
        acc[r] = 0.0f;
    }
    __syncthreads();

    // Stream j over the strip; every lane reads the same LDS word (broadcast).
    // Full-matrix accumulation (i==j contributes exactly 0).
#pragma unroll 4
    for (int jj = 0; jj < TAU_TPB; ++jj) {
        const float pj = sp[jj];
        const float lj = sl[jj];
#pragma unroll
        for (int r = 0; r < TAU_IPT; ++r) {
            const float t  = fast_tanhf(pi[r] - pj);
            const float ld = li[r] - lj;
            // sign(ld) * t without a multiply: two v_cndmask
            acc[r] += (ld > 0.0f) ? t : ((ld < 0.0f) ? -t : 0.0f);
        }
    }

    float tsum = (acc[0] + acc[1]) + (acc[2] + acc[3]);

    // Deterministic block tree-reduction in LDS.
    __shared__ float red[TAU_TPB];
    red[tid] = tsum;
    __syncthreads();
#pragma unroll
    for (int s = TAU_TPB / 2; s > 0; s >>= 1) {
        if (tid < s) red[tid] += red[tid + s];
        __syncthreads();
    }
    if (tid == 0)
        partial[(size_t)js * TAU_C + c] = red[0];
}

// ---------------------------------------------------------------------------
// Kernel 2: one wave32 reduces P=512 partials and writes the scalar loss.
// Uses V_WMMA_F32_16X16X4_F32 with B = ones to fold 32 lane sums:
//   A[m][0]=lanesum[m] (lanes 0..15, VGPR0), A[m][2]=lanesum[m+16] (lanes 16..31)
//   => D[m][n] = lanesum[m] + lanesum[m+16]
//   D column 0 lives in d[0..7] of lane 0 (M=0..7) and lane 16 (M=8..15).
// ---------------------------------------------------------------------------
__global__ __launch_bounds__(32)
void tau_finalize_kernel(const float* __restrict__ partial,
                         float* __restrict__ out, int P) {
    const int lane = threadIdx.x;        // 0..31, no divergence before WMMA
    float s = 0.0f;
    for (int k = lane; k < P; k += 32)   // P multiple of 32 -> uniform trip count
        s += partial[k];

    float total;
#if __has_builtin(__builtin_amdgcn_wmma_f32_16x16x4_f32)
    v2f a; a.x = s;    a.y = 0.0f;       // A: K0/K2 = lane sums, K1/K3 = 0
    v2f b; b.x = 1.0f; b.y = 1.0f;       // B: all ones (4x16)
    v8f cz = {};
    v8f d = __builtin_amdgcn_wmma_f32_16x16x4_f32(
        /*neg_a=*/false, a, /*neg_b=*/false, b,
        /*c_mod=*/(short)0, cz, /*reuse_a=*/false, /*reuse_b=*/false);
    float rowsum = ((d[0] + d[1]) + (d[2] + d[3])) +
                   ((d[4] + d[5]) + (d[6] + d[7]));
    // lane 0 holds sum of D[0..7][0], lane 16 holds sum of D[8..15][0]
    total = __shfl(rowsum, 0, 32) + __shfl(rowsum, 16, 32);
#else
    // Fallback: shuffle tree across the wave.
    float v = s;
    for (int off = 16; off > 0; off >>= 1)
        v += __shfl_down(v, off, 32);
    total = __shfl(v, 0, 32);
#endif

    if (lane == 0) {
        const float denom = (float)TAU_N * (float)(TAU_N - 1) * (float)TAU_C;
        out[0] = 1.0f - total / denom;
    }
}

// ---------------------------------------------------------------------------
extern "C" void kernel_launch(void* const* d_in, const int* in_sizes, int n_in,
                              void* d_out, int out_size, void* d_ws, size_t ws_size,
                              hipStream_t stream) {
    const float* pred = (const float*)d_in[0];   // [128, 1024] fp32
    const float* y    = (const float*)d_in[1];   // [128, 1024] fp32
    float* out        = (float*)d_out;           // scalar fp32
    float* partial    = (float*)d_ws;            // 512 floats of scratch

    dim3 grid(TAU_C, TAU_JSTRIPS);               // 128 x 4 = 512 blocks
    tau_partial_kernel<<<grid, TAU_TPB, 0, stream>>>(pred, y, partial);

    tau_finalize_kernel<<<1, 32, 0, stream>>>(partial, out,
                                              TAU_C * TAU_JSTRIPS);
}